// RandomCrop_39539468927487
// MI455X (gfx1250) — compile-verified
//
#include <hip/hip_runtime.h>
#include <stdint.h>

// RandomCrop: B=64 images [512,512,3] f32 -> per-image dynamic 448x448x3 crop.
// Pure bandwidth op (~308 MB HBM traffic -> ~13 us at 23.3 TB/s).
// Strategy: Tensor Data Mover 2D tile copy, global -> LDS -> global.

#define IMG_H 512
#define IMG_W 512
#define CH    3
#define CROP  448
#define SRC_ROW (IMG_W * CH)   // 1536 elements per source row
#define DST_ROW (CROP * CH)    // 1344 elements per crop row
#define ROWS_PER_SLAB 8
#define SLABS (CROP / ROWS_PER_SLAB)   // 56 slabs per image

typedef unsigned int v4u __attribute__((ext_vector_type(4)));
typedef int          v8i __attribute__((ext_vector_type(8)));
typedef int          v4i __attribute__((ext_vector_type(4)));

#if defined(__AMDGCN__) && __has_builtin(__builtin_amdgcn_tensor_load_to_lds) && \
    __has_builtin(__builtin_amdgcn_tensor_store_from_lds) && \
    __has_builtin(__builtin_amdgcn_s_wait_tensorcnt)
#define USE_TDM 1
#else
#define USE_TDM 0
#endif

__global__ __launch_bounds__(32)
void crop_tdm_kernel(const float* __restrict__ src,
                     const int* __restrict__ idx_h,
                     const int* __restrict__ idx_w,
                     float* __restrict__ dst) {
    const int img  = blockIdx.x / SLABS;
    const int slab = blockIdx.x % SLABS;
    const int h0 = idx_h[img];
    const int w0 = idx_w[img];

#if USE_TDM
    __shared__ float lds_buf[ROWS_PER_SLAB * DST_ROW];   // 43008 bytes

    // Low 32 bits of a flat LDS pointer are the LDS byte offset.
    const uint32_t lds_off = (uint32_t)(uintptr_t)&lds_buf[0];

    const uint64_t src_addr = (uint64_t)(uintptr_t)src +
        4ull * ((uint64_t)img * (IMG_H * IMG_W * CH) +
                (uint64_t)(h0 + slab * ROWS_PER_SLAB) * SRC_ROW +
                (uint64_t)w0 * CH);
    const uint64_t dst_addr = (uint64_t)(uintptr_t)dst +
        4ull * ((uint64_t)img * (CROP * DST_ROW) +
                (uint64_t)(slab * ROWS_PER_SLAB) * DST_ROW);

    // ---- D# group 0 (load: global tile -> LDS) ----
    // [1:0]=count=1, [63:32]=lds_addr, [120:64]=global_addr, [127:126]=type=2
    v4u g0_ld;
    g0_ld.x = 1u;
    g0_ld.y = lds_off;
    g0_ld.z = (uint32_t)src_addr;
    g0_ld.w = (uint32_t)((src_addr >> 32) & 0x1FFFFFFu) | (2u << 30);

    // ---- D# group 1 (load) ----
    // [17:16]=data_size(2 -> 4B); [79:48]=tensor_dim0; [111:80]=tensor_dim1;
    // [127:112]=tile_dim0; [143:128]=tile_dim1; [159:144]=tile_dim2(0);
    // [207:160]=tensor_dim0_stride; [255:208]=tensor_dim1_stride(unused, 0)
    v8i g1_ld;
    g1_ld[0] = (int)(2u << 16);                        // data_size = 4 bytes
    g1_ld[1] = (int)((uint32_t)SRC_ROW << 16);         // tensor_dim0[15:0] = 1536
    g1_ld[2] = (int)((uint32_t)IMG_H   << 16);         // tensor_dim1[15:0] = 512
    g1_ld[3] = (int)((uint32_t)DST_ROW << 16);         // tile_dim0 = 1344
    g1_ld[4] = ROWS_PER_SLAB;                          // tile_dim1 = 8, tile_dim2 = 0
    g1_ld[5] = SRC_ROW;                                // tensor_dim0_stride = 1536
    g1_ld[6] = 0;
    g1_ld[7] = 0;

    // ---- D# for store (LDS -> contiguous global dst) ----
    v4u g0_st = g0_ld;
    g0_st.z = (uint32_t)dst_addr;
    g0_st.w = (uint32_t)((dst_addr >> 32) & 0x1FFFFFFu) | (2u << 30);

    v8i g1_st = g1_ld;
    g1_st[1] = (int)((uint32_t)DST_ROW << 16);         // tensor_dim0 = 1344
    g1_st[2] = (int)((uint32_t)CROP    << 16);         // tensor_dim1 = 448
    g1_st[5] = DST_ROW;                                // stride = 1344 (contiguous)

    v4i z4 = {0, 0, 0, 0};                             // groups 2/3 unused (2D tensor)
    v8i z8 = {0, 0, 0, 0, 0, 0, 0, 0};                 // extra arg on clang-23 form

    __builtin_amdgcn_tensor_load_to_lds(g0_ld, g1_ld, z4, z4, z8, 0);
    __builtin_amdgcn_s_wait_tensorcnt(0);              // tile resident in LDS
    __builtin_amdgcn_tensor_store_from_lds(g0_st, g1_st, z4, z4, z8, 0);
    __builtin_amdgcn_s_wait_tensorcnt(0);              // drain before wave exit
#else
    // Fallback: coalesced copy. Source rows are only 4B aligned (w0*12 byte
    // offset), so gather 4 scalars and store one aligned float4.
    const float* s0 = src + (size_t)img * (IMG_H * IMG_W * CH)
                          + (size_t)(h0 + slab * ROWS_PER_SLAB) * SRC_ROW
                          + (size_t)w0 * CH;
    float* dbase = dst + (size_t)img * (CROP * DST_ROW)
                       + (size_t)slab * ROWS_PER_SLAB * DST_ROW;
    const int n4 = ROWS_PER_SLAB * DST_ROW / 4;        // 2688 float4 per slab
    for (int i = threadIdx.x; i < n4; i += 32) {
        int e = i * 4;
        int row = e / DST_ROW;
        int col = e - row * DST_ROW;
        const float* s = s0 + (size_t)row * SRC_ROW + col;
        float4 v = { s[0], s[1], s[2], s[3] };
        ((float4*)dbase)[i] = v;
    }
#endif
}

extern "C" void kernel_launch(void* const* d_in, const int* in_sizes, int n_in,
                              void* d_out, int out_size, void* d_ws, size_t ws_size,
                              hipStream_t stream) {
    const float* images = (const float*)d_in[0];
    const int*   idx_h  = (const int*)d_in[1];
    const int*   idx_w  = (const int*)d_in[2];
    float*       out    = (float*)d_out;

    const int B = in_sizes[1];              // 64 images (idx_h has B entries)
    dim3 grid(B * SLABS);                   // one (image, 8-row slab) per wave
    crop_tdm_kernel<<<grid, 32, 0, stream>>>(images, idx_h, idx_w, out);
}